// GraphConvolution_50190987821615
// MI455X (gfx1250) — compile-verified
//
#include <hip/hip_runtime.h>
#include <hip/hip_bf16.h>

// GraphConvolution: agg = relu( SpMM(A, x @ W^T + b) )
//   x:[N,128] f32, edges (row,col,val):[E], W:[128,128] f32, b:[128] f32
// Pipeline (all on `stream`):
//   1) zero d_out (atomic accumulation target; harness does not re-zero)
//   2) WMMA fp32 GEMM: lin = x @ W^T + b   -> d_ws
//   3) edge-parallel SpMM with global f32 atomics into d_out
//   4) in-place ReLU on d_out

typedef float v2f __attribute__((ext_vector_type(2)));
typedef float v8f __attribute__((ext_vector_type(8)));

static constexpr int DIM = 128;         // D_IN == D_OUT == 128

// ---------------------------------------------------------------------------
// 1) zero-fill (float4 grid-stride)
// ---------------------------------------------------------------------------
__global__ void gc_zero_kernel(float4* __restrict__ p, int n4) {
    int i = blockIdx.x * blockDim.x + threadIdx.x;
    int stride = gridDim.x * blockDim.x;
    float4 z = make_float4(0.f, 0.f, 0.f, 0.f);
    for (; i < n4; i += stride) p[i] = z;
}

// ---------------------------------------------------------------------------
// 2) fp32 WMMA GEMM: lin[n][o] = sum_k x[n][k] * W[o][k] + b[o]
//    One wave computes a 16-row x 128-col strip using V_WMMA_F32_16X16X4_F32.
//    A frag (16x4 f32): lane L -> row (L&15), VGPR v -> k = (L>>4)*2 + v
//    B frag (4x16 f32): lane L -> col (L&15), VGPR v -> k = (L>>4)*2 + v
//    C/D (16x16 f32):   lane L, VGPR v -> row (L>>4)*8 + v, col (L&15)
// ---------------------------------------------------------------------------
__global__ __launch_bounds__(256)
void gc_gemm_wmma_kernel(const float* __restrict__ x,
                         const float* __restrict__ W,
                         const float* __restrict__ b,
                         float* __restrict__ lin,
                         int nNodes) {
    const int lane      = threadIdx.x & 31;
    const int waveId    = threadIdx.x >> 5;                 // 8 waves / block
    const int rowTile   = blockIdx.x * 8 + waveId;
    const int numTiles  = (nNodes + 15) >> 4;
    if (rowTile >= numTiles) return;                        // wave-uniform: EXEC stays all-1s

    const int rowBase = rowTile << 4;
    const int ln15    = lane & 15;                          // A row / B col / D col
    const int kh      = (lane >> 4) << 1;                   // 0 or 2: K sub-offset
    const int half    = lane >> 4;

    // accumulators seeded with bias (bias depends only on output column)
    v8f acc[8];
#pragma unroll
    for (int nt = 0; nt < 8; ++nt) {
        const float bv = b[nt * 16 + ln15];
#pragma unroll
        for (int v = 0; v < 8; ++v) acc[nt][v] = bv;
    }

    int ra = rowBase + ln15;
    if (ra >= nNodes) ra = nNodes - 1;                      // clamp tail reads
    const float* __restrict__ xrow = x + (size_t)ra * DIM;

#pragma unroll 1
    for (int kk = 0; kk < DIM; kk += 4) {
        v2f a;
        a.x = xrow[kk + kh + 0];
        a.y = xrow[kk + kh + 1];
#pragma unroll
        for (int nt = 0; nt < 8; ++nt) {
            const float* __restrict__ wrow =
                W + (size_t)(nt * 16 + ln15) * DIM + kk + kh;
            v2f bf;
            bf.x = wrow[0];
            bf.y = wrow[1];
            // All 8 WMMAs of this K-step share the same A fragment: set the
            // reuse-A OPSEL hint on instructions 2..8 (identical to previous).
            if (nt == 0) {
                acc[nt] = __builtin_amdgcn_wmma_f32_16x16x4_f32(
                    false, a, false, bf, (short)0, acc[nt],
                    /*reuse_a=*/false, /*reuse_b=*/false);
            } else {
                acc[nt] = __builtin_amdgcn_wmma_f32_16x16x4_f32(
                    false, a, false, bf, (short)0, acc[nt],
                    /*reuse_a=*/true, /*reuse_b=*/false);
            }
        }
    }

    // store D: lane L, VGPR v -> row (L>>4)*8+v, col nt*16 + (L&15)
    float* __restrict__ dbase = lin + (size_t)(rowBase + half * 8) * DIM + ln15;
    if (rowBase + 16 <= nNodes) {
        // fast path: whole tile in bounds, unguarded claused stores
#pragma unroll
        for (int v = 0; v < 8; ++v) {
#pragma unroll
            for (int nt = 0; nt < 8; ++nt) {
                dbase[(size_t)v * DIM + nt * 16] = acc[nt][v];
            }
        }
    } else {
        // ragged tail tile (only when nNodes % 16 != 0)
#pragma unroll
        for (int v = 0; v < 8; ++v) {
            const int row = rowBase + half * 8 + v;
            if (row < nNodes) {
#pragma unroll
                for (int nt = 0; nt < 8; ++nt) {
                    dbase[(size_t)v * DIM + nt * 16] = acc[nt][v];
                }
            }
        }
    }
}

// ---------------------------------------------------------------------------
// 3) SpMM scatter: one wave per edge; lane handles 4 consecutive features.
//    agg[r][:] += val * lin[c][:]   via global_atomic_add_f32
// ---------------------------------------------------------------------------
__global__ __launch_bounds__(256)
void gc_spmm_kernel(const int* __restrict__ erow,
                    const int* __restrict__ ecol,
                    const float* __restrict__ eval_,
                    const float* __restrict__ lin,
                    float* __restrict__ agg,
                    int nEdges) {
    const int edge = (blockIdx.x * blockDim.x + threadIdx.x) >> 5;
    const int lane = threadIdx.x & 31;
    if (edge >= nEdges) return;                             // wave-uniform

    const int   r = erow[edge];
    const int   c = ecol[edge];
    const float v = eval_[edge];

    const float4 m = *((const float4*)(lin + (size_t)c * DIM) + lane);
    float* dst = agg + (size_t)r * DIM + lane * 4;
    atomicAdd(dst + 0, v * m.x);
    atomicAdd(dst + 1, v * m.y);
    atomicAdd(dst + 2, v * m.z);
    atomicAdd(dst + 3, v * m.w);
}

// ---------------------------------------------------------------------------
// 4) in-place ReLU (float4)
// ---------------------------------------------------------------------------
__global__ void gc_relu_kernel(float4* __restrict__ p, int n4) {
    int i = blockIdx.x * blockDim.x + threadIdx.x;
    if (i >= n4) return;
    float4 t = p[i];
    t.x = t.x > 0.f ? t.x : 0.f;
    t.y = t.y > 0.f ? t.y : 0.f;
    t.z = t.z > 0.f ? t.z : 0.f;
    t.w = t.w > 0.f ? t.w : 0.f;
    p[i] = t;
}

// ---------------------------------------------------------------------------
extern "C" void kernel_launch(void* const* d_in, const int* in_sizes, int n_in,
                              void* d_out, int out_size, void* d_ws, size_t ws_size,
                              hipStream_t stream) {
    const float* x     = (const float*)d_in[0];
    const int*   erow  = (const int*)  d_in[1];
    const int*   ecol  = (const int*)  d_in[2];
    const float* eval_ = (const float*)d_in[3];
    const float* W     = (const float*)d_in[4];
    const float* b     = (const float*)d_in[5];

    float* agg = (float*)d_out;          // [N,128] accumulation target
    float* lin = (float*)d_ws;           // [N,128] linear output scratch

    const int nNodes = in_sizes[0] / DIM;
    const int nEdges = in_sizes[1];
    const int nElems = nNodes * DIM;
    const int n4     = (nElems + 3) / 4;

    // 1) zero the accumulation buffer (harness poisons, never re-zeros)
    {
        int blocks = (n4 + 255) / 256;
        if (blocks > 65535) blocks = 65535;                 // grid-stride
        gc_zero_kernel<<<blocks, 256, 0, stream>>>((float4*)agg, n4);
    }

    // 2) WMMA GEMM -> lin
    {
        const int numTiles = (nNodes + 15) / 16;
        const int blocks   = (numTiles + 7) / 8;            // 8 waves/block
        gc_gemm_wmma_kernel<<<blocks, 256, 0, stream>>>(x, W, b, lin, nNodes);
    }

    // 3) edge-parallel SpMM with f32 atomics (8 edges per 256-thread block)
    {
        const int blocks = (nEdges + 7) / 8;
        gc_spmm_kernel<<<blocks, 256, 0, stream>>>(erow, ecol, eval_, lin, agg, nEdges);
    }

    // 4) ReLU in place
    {
        const int blocks = (n4 + 255) / 256;
        gc_relu_kernel<<<blocks, 256, 0, stream>>>((float4*)agg, n4);
    }
}